// DCGCNEncoder_28578712388230
// MI455X (gfx1250) — compile-verified
//
#include <hip/hip_runtime.h>
#include <hip/hip_bf16.h>

typedef __attribute__((ext_vector_type(2))) float v2f;
typedef __attribute__((ext_vector_type(8))) float v8f;

// ---------------------------------------------------------------------------
// Degree / normalization kernels
// ---------------------------------------------------------------------------
__global__ void k_fill1(float* __restrict__ p, int n) {
  int i = blockIdx.x * blockDim.x + threadIdx.x;
  if (i < n) p[i] = 1.0f;                       // self-loop contributes 1 to deg
}

__global__ void k_degcount(const int* __restrict__ ei, int E, float* __restrict__ deg) {
  int e = blockIdx.x * blockDim.x + threadIdx.x;
  if (e < E) atomicAdd(&deg[ei[E + e]], 1.0f);  // dst row of edge_index
}

__global__ void k_rsqrt_inplace(float* __restrict__ p, int n) {
  int i = blockIdx.x * blockDim.x + threadIdx.x;
  if (i < n) p[i] = rsqrtf(p[i]);               // deg >= 1 always (self-loops)
}

// ---------------------------------------------------------------------------
// h = (optional relu(x)) @ W   via V_WMMA_F32_16X16X4_F32, fp32 exact.
// One wave computes one 16x16 tile of h. K stepped by 4.
// A frag (16x4 f32): lanes 0-15 -> K={k0,k0+1}, lanes 16-31 -> K={k0+2,k0+3}
// B frag (4x16 f32): mirrored over N. C/D: 8 VGPRs, lanes 0-15 = N, vgpr r ->
// M=r (low half) / M=r+8 (high half).
// ---------------------------------------------------------------------------
template <int K, int F, bool RELU_IN>
__global__ void k_gemm_wmma(const float* __restrict__ x, const float* __restrict__ W,
                            float* __restrict__ h, int rowTiles) {
  constexpr int colTiles = F / 16;
  const int wavesPerBlock = blockDim.x >> 5;
  const int wv = blockIdx.x * wavesPerBlock + (threadIdx.x >> 5);
  if (wv >= rowTiles * colTiles) return;        // wave-uniform: EXEC stays all-ones
  const int tm   = wv / colTiles;
  const int tn   = wv % colTiles;
  const int lane = threadIdx.x & 31;
  const int half = lane >> 4;                   // which K-pair this half-wave owns
  const int l    = lane & 15;
  const int row  = tm * 16 + l;
  const int col  = tn * 16 + l;

  v8f acc = {0.f, 0.f, 0.f, 0.f, 0.f, 0.f, 0.f, 0.f};
#pragma unroll
  for (int k0 = 0; k0 < K; k0 += 4) {
    const int ka = k0 + half * 2;
    float a0 = x[row * K + ka + 0];
    float a1 = x[row * K + ka + 1];
    if (RELU_IN) { a0 = fmaxf(a0, 0.f); a1 = fmaxf(a1, 0.f); }
    v2f a = {a0, a1};
    v2f b = {W[(ka + 0) * F + col], W[(ka + 1) * F + col]};
    // 8 args: (neg_a, A, neg_b, B, c_mod, C, reuse_a, reuse_b)
    acc = __builtin_amdgcn_wmma_f32_16x16x4_f32(false, a, false, b, (short)0, acc,
                                                false, false);
  }
#pragma unroll
  for (int r = 0; r < 8; ++r)
    h[(tm * 16 + r + half * 8) * F + col] = acc[r];
}

// ---------------------------------------------------------------------------
// out[i][f] = dinv[i]^2 * h[i][f] + b[f]   (self-loop message + bias; also
// overwrites the 0xAA poison in d_out before the atomic scatter begins)
// ---------------------------------------------------------------------------
template <int F>
__global__ void k_init_out(const float* __restrict__ h, const float* __restrict__ dinv,
                           const float* __restrict__ b, float* __restrict__ out, int n) {
  int t = blockIdx.x * blockDim.x + threadIdx.x;
  if (t < n * F) {
    const int i = t / F;
    const int f = t % F;
    const float di = dinv[i];
    out[t] = di * di * h[t] + b[f];
  }
}

// ---------------------------------------------------------------------------
// Scatter-add of edge messages: one thread per (edge, feature).
// Gather hits L2 (h is ~12.8MB << 192MB L2); atomicAdd lowers to a
// non-returning GLOBAL_ATOMIC_ADD_F32 (fire-and-forget, STOREcnt).
// Wave accesses are contiguous 128B runs within a node's feature row.
// ---------------------------------------------------------------------------
template <int F>
__global__ void k_scatter(const int* __restrict__ ei, int E,
                          const float* __restrict__ h, const float* __restrict__ dinv,
                          float* __restrict__ out) {
  const unsigned t = blockIdx.x * blockDim.x + threadIdx.x;
  const unsigned total = (unsigned)E * F;
  if (t >= total) return;
  const int e = t / F;
  const int f = t % F;
  const int s = ei[e];
  const int d = ei[E + e];
  const float norm = dinv[s] * dinv[d];
  atomicAdd(&out[d * F + f], norm * h[s * F + f]);
}

__global__ void k_relu_inplace(float* __restrict__ p, int n) {
  int i = blockIdx.x * blockDim.x + threadIdx.x;
  if (i < n) p[i] = fmaxf(p[i], 0.f);
}

// ---------------------------------------------------------------------------
// Launch
// ---------------------------------------------------------------------------
static inline int cdiv(long long a, int b) { return (int)((a + b - 1) / b); }

extern "C" void kernel_launch(void* const* d_in, const int* in_sizes, int n_in,
                              void* d_out, int out_size, void* d_ws, size_t ws_size,
                              hipStream_t stream) {
  const float* x0 = (const float*)d_in[0];
  const int*   e1 = (const int*)d_in[1];
  const int*   e3 = (const int*)d_in[2];
  const int*   e9 = (const int*)d_in[3];
  const float* W1 = (const float*)d_in[4];
  const float* b1 = (const float*)d_in[5];
  const float* W2 = (const float*)d_in[6];
  const float* b2 = (const float*)d_in[7];
  const float* W3 = (const float*)d_in[8];
  const float* b3 = (const float*)d_in[9];

  const int N = in_sizes[0] / 64;   // 50000 (== 3125 * 16, tile-exact)
  const int E = in_sizes[1] / 2;    // 800000

  // workspace carve (floats): dinv | h | x1 | x2
  float* dinv = (float*)d_ws;
  float* h    = dinv + N;
  float* x1   = h  + (size_t)N * 64;
  float* x2   = x1 + (size_t)N * 64;
  float* out  = (float*)d_out;

  const int rowTiles = N / 16;
  const int B = 256;                 // 8 waves/block (wave32)

  // ---------------- Layer 1: Fin=64 -> Fout=64, edges e1, input x0 (no relu)
  k_fill1<<<cdiv(N, B), B, 0, stream>>>(dinv, N);
  k_degcount<<<cdiv(E, B), B, 0, stream>>>(e1, E, dinv);
  k_rsqrt_inplace<<<cdiv(N, B), B, 0, stream>>>(dinv, N);
  k_gemm_wmma<64, 64, false><<<cdiv(rowTiles * 4, 8), B, 0, stream>>>(x0, W1, h, rowTiles);
  k_init_out<64><<<cdiv((long long)N * 64, B), B, 0, stream>>>(h, dinv, b1, x1, N);
  k_scatter<64><<<cdiv((long long)E * 64, B), B, 0, stream>>>(e1, E, h, dinv, x1);
  // relu folded into layer-2 GEMM input load

  // ---------------- Layer 2: Fin=64 -> Fout=32, edges e3, input relu(x1)
  k_fill1<<<cdiv(N, B), B, 0, stream>>>(dinv, N);
  k_degcount<<<cdiv(E, B), B, 0, stream>>>(e3, E, dinv);
  k_rsqrt_inplace<<<cdiv(N, B), B, 0, stream>>>(dinv, N);
  k_gemm_wmma<64, 32, true><<<cdiv(rowTiles * 2, 8), B, 0, stream>>>(x1, W2, h, rowTiles);
  k_init_out<32><<<cdiv((long long)N * 32, B), B, 0, stream>>>(h, dinv, b2, x2, N);
  k_scatter<32><<<cdiv((long long)E * 32, B), B, 0, stream>>>(e3, E, h, dinv, x2);

  // ---------------- Layer 3: Fin=32 -> Fout=16, edges e9, input relu(x2)
  k_fill1<<<cdiv(N, B), B, 0, stream>>>(dinv, N);
  k_degcount<<<cdiv(E, B), B, 0, stream>>>(e9, E, dinv);
  k_rsqrt_inplace<<<cdiv(N, B), B, 0, stream>>>(dinv, N);
  k_gemm_wmma<32, 16, true><<<cdiv(rowTiles * 1, 8), B, 0, stream>>>(x2, W3, h, rowTiles);
  k_init_out<16><<<cdiv((long long)N * 16, B), B, 0, stream>>>(h, dinv, b3, out, N);
  k_scatter<16><<<cdiv((long long)E * 16, B), B, 0, stream>>>(e9, E, h, dinv, out);
  k_relu_inplace<<<cdiv((long long)N * 16, B), B, 0, stream>>>(out, N * 16);
}